// Cam2Cam_8718783611170
// MI455X (gfx1250) — compile-verified
//
#include <hip/hip_runtime.h>

#define BATCH 16
#define HH 480
#define WW 640
#define HW (HH * WW)
#define EPSF 1e-7f

typedef __attribute__((ext_vector_type(2))) float v2f;
typedef __attribute__((ext_vector_type(8))) float v8f;

__global__ __launch_bounds__(256) void cam2cam_kernel(
    const float* __restrict__ img,    // B,3,H,W
    const float* __restrict__ depth,  // B,1,H,W
    const float* __restrict__ T,      // B,4,4
    const float* __restrict__ K,      // B,4,4
    const float* __restrict__ invK,   // B,4,4
    float* __restrict__ out)          // B,3,H,W
{
    const int b    = blockIdx.y;
    const int p    = blockIdx.x * 256 + threadIdx.x;   // pixel id in batch (exact cover)
    const int lane = threadIdx.x & 31;
    const int py   = p / WW;
    const int px   = p - py * WW;

    const float* Kb = K    + b * 16;
    const float* Tb = T    + b * 16;
    const float* Ib = invK + b * 16;

    // ---- back-project: cam = depth * invK[:3,:3] @ (x, y, 1); cam3 = 1 ----
    const float x = (float)px;
    const float y = (float)py;
    const float d = __builtin_nontemporal_load(depth + (size_t)b * HW + p);
    const float c0 = d * (Ib[0] * x + Ib[1] * y + Ib[2]);
    const float c1 = d * (Ib[4] * x + Ib[5] * y + Ib[6]);
    const float c2 = d * (Ib[8] * x + Ib[9] * y + Ib[10]);
    // c3 == 1.0f

    // ---- build A operand: P = (K@T)[:3,:] placed in rows {0,1,2} and {8,9,10} ----
    // A layout (32-bit 16x4): lanes 0-15 -> M=lane, VGPR0=K0, VGPR1=K1;
    //                         lanes 16-31 -> M=lane-16, VGPR0=K2, VGPR1=K3.
    const int row   = lane & 15;
    const int er    = row & 7;                 // P row for M in {0,1,2,8,9,10}
    const float val = (er < 3) ? 1.0f : 0.0f;
    const int er3   = (er < 3) ? er : 0;       // keep loads in-range
    const int kb0   = (lane < 16) ? 0 : 2;     // K column pair this half-wave holds
    const int r4    = er3 * 4;
    const float p0 = Kb[r4 + 0] * Tb[0 * 4 + kb0] + Kb[r4 + 1] * Tb[1 * 4 + kb0] +
                     Kb[r4 + 2] * Tb[2 * 4 + kb0] + Kb[r4 + 3] * Tb[3 * 4 + kb0];
    const float p1 = Kb[r4 + 0] * Tb[0 * 4 + kb0 + 1] + Kb[r4 + 1] * Tb[1 * 4 + kb0 + 1] +
                     Kb[r4 + 2] * Tb[2 * 4 + kb0 + 1] + Kb[r4 + 3] * Tb[3 * 4 + kb0 + 1];
    v2f Av;
    Av[0] = val * p0;
    Av[1] = val * p1;

    // ---- build B operands: B[k][n], lanes 0-15 -> K=0/1, lanes 16-31 -> K=2/3, N=lane&15 ----
    // B1 = cam of wave pixels 0..15, B2 = cam of wave pixels 16..31.
    const float c0s = __shfl_xor(c0, 16, 32);
    const float c1s = __shfl_xor(c1, 16, 32);
    const float c2s = __shfl_xor(c2, 16, 32);
    const bool lo = (lane < 16);
    v2f B1, B2;
    B1[0] = lo ? c0  : c2s;  B1[1] = lo ? c1  : 1.0f;
    B2[0] = lo ? c0s : c2;   B2[1] = lo ? c1s : 1.0f;

    // ---- cp = P @ cam via WMMA (two 16-pixel groups) ----
    v8f zero = {};
    v8f d1 = __builtin_amdgcn_wmma_f32_16x16x4_f32(false, Av, false, B1,
                                                   (short)0, zero, false, false);
    v8f d2 = __builtin_amdgcn_wmma_f32_16x16x4_f32(false, Av, false, B2,
                                                   (short)0, zero, false, false);
    // D layout: lanes 0-15 read M=0,1,2 (VGPR 0..2) at N=lane;
    //           lanes 16-31 read M=8,9,10 (VGPR 0..2) at N=lane-16 -> own pixel via row dup.
    const float cpx = lo ? d1[0] : d2[0];
    const float cpy = lo ? d1[1] : d2[1];
    const float cpz = lo ? d1[2] : d2[2];

    // ---- perspective divide + grid coords, algebraically folded ----
    // gx = (px2/(W-1) - 0.5)*2 ; ix = ((gx+1)*W - 1)*0.5  ==>  ix = px2*(W/(W-1)) - 0.5
    const float iz = cpz + EPSF;
    float rz = __builtin_amdgcn_rcpf(iz);              // ~1 ulp hw reciprocal
    rz = fmaf(fmaf(-iz, rz, 1.0f), rz, rz);            // one Newton step -> ~0.5 ulp
    const float rzx = rz * ((float)WW / (float)(WW - 1));
    const float rzy = rz * ((float)HH / (float)(HH - 1));
    const float ix = fmaf(cpx, rzx, -0.5f);
    const float iy = fmaf(cpy, rzy, -0.5f);

    // ---- bilinear sample with border clamp (float-domain clamps match reference) ----
    const float ix0 = floorf(ix);
    const float iy0 = floorf(iy);
    const float wx = ix - ix0;
    const float wy = iy - iy0;
    const int ix0c = (int)fminf(fmaxf(ix0,        0.0f), (float)(WW - 1));
    const int ix1c = (int)fminf(fmaxf(ix0 + 1.0f, 0.0f), (float)(WW - 1));
    const int iy0c = (int)fminf(fmaxf(iy0,        0.0f), (float)(HH - 1));
    const int iy1c = (int)fminf(fmaxf(iy0 + 1.0f, 0.0f), (float)(HH - 1));

    const float w00 = (1.0f - wx) * (1.0f - wy);
    const float w01 = wx * (1.0f - wy);
    const float w10 = (1.0f - wx) * wy;
    const float w11 = wx * wy;

    const int i00 = iy0c * WW + ix0c;
    const int i01 = iy0c * WW + ix1c;
    const int i10 = iy1c * WW + ix0c;
    const int i11 = iy1c * WW + ix1c;

    const float* imb = img + (size_t)b * 3 * HW;
    float*       ob  = out + (size_t)b * 3 * HW;
#pragma unroll
    for (int ch = 0; ch < 3; ++ch) {
        const float* ic = imb + (size_t)ch * HW;
        const float v = ic[i00] * w00 + ic[i01] * w01 + ic[i10] * w10 + ic[i11] * w11;
        __builtin_nontemporal_store(v, &ob[(size_t)ch * HW + p]);
    }
}

extern "C" void kernel_launch(void* const* d_in, const int* in_sizes, int n_in,
                              void* d_out, int out_size, void* d_ws, size_t ws_size,
                              hipStream_t stream) {
    const float* img   = (const float*)d_in[0];
    const float* depth = (const float*)d_in[1];
    const float* T     = (const float*)d_in[2];
    const float* K     = (const float*)d_in[3];
    const float* invK  = (const float*)d_in[4];
    float* out = (float*)d_out;

    dim3 grid(HW / 256, BATCH);  // 307200 % 256 == 0 -> full waves, EXEC all-1s for WMMA
    cam2cam_kernel<<<grid, dim3(256), 0, stream>>>(img, depth, T, K, invK, out);
}